// LocalAffineAlignment_55679956025825
// MI455X (gfx1250) — compile-verified
//
#include <hip/hip_runtime.h>

#ifndef __has_builtin
#define __has_builtin(x) 0
#endif

namespace {
constexpr int kB = 16, kC = 3, kH = 512, kW = 640;
constexpr int kP = 8;
constexpr int kNH = kH / kP;       // 64 patch rows
constexpr int kWavesPerBlk = 10;   // 10 waves x 32 cols = 320 cols per block
constexpr int kBlkCols = 32 * kWavesPerBlk;  // 320
constexpr float kEPS = 1e-6f;
constexpr float kMinValid = 6.0f;  // int(0.1 * 8 * 8)
} // namespace

typedef int v4i __attribute__((vector_size(4 * sizeof(int))));
typedef __attribute__((address_space(1))) int*  gint_p;
typedef __attribute__((address_space(3))) int*  lint_p;
typedef __attribute__((address_space(1))) v4i*  gv4i_p;
typedef __attribute__((address_space(3))) v4i*  lv4i_p;

__device__ __forceinline__ float red8(float v) {
  // xor-butterfly sum over each aligned group of 8 lanes (one 8x8 patch).
  // Leaves the full patch sum replicated in all 8 lanes of the group.
  v += __shfl_xor(v, 1, 32);
  v += __shfl_xor(v, 2, 32);
  v += __shfl_xor(v, 4, 32);
  return v;
}

__global__ __launch_bounds__(kBlkCols) void laa_fused(
    const float* __restrict__ T, const float* __restrict__ S,
    const float* __restrict__ V, float* __restrict__ oA,
    float* __restrict__ oAl, float* __restrict__ oBe)
{
  // per-wave mask tile: 8 rows x 32 cols
  __shared__ float smask[kWavesPerBlk][8 * 32];

  const int lane = threadIdx.x & 31;
  const int wib  = threadIdx.x >> 5;

  // 3D grid: no integer division anywhere.
  const int b  = blockIdx.z;                    // batch
  const int gy = blockIdx.y;                    // patch row
  const int sx = blockIdx.x * kBlkCols + wib * 32;  // wave strip base column
  const int x  = sx + lane;
  const int y0 = gy * kP;

  const size_t mBase = (size_t)b * kH * kW;       // mask / alpha / beta plane
  const size_t iBase = (size_t)b * kC * kH * kW;  // image base

  // ---- stage validity mask via async global->LDS (ASYNCcnt-tracked DMA) ----
#if __has_builtin(__builtin_amdgcn_global_load_async_to_lds_b128)
  {
    // lane -> (row = lane>>3, 16B column group = (lane&7)*4); 2 issues cover 8 rows
    const int r0 = lane >> 3;
    const int cg = (lane & 7) * 4;
    #pragma unroll
    for (int p = 0; p < 2; ++p) {
      __builtin_amdgcn_global_load_async_to_lds_b128(
          (gv4i_p)(V + mBase + (size_t)(y0 + p * 4 + r0) * kW + sx + cg),
          (lv4i_p)&smask[wib][(p * 4 + r0) * 32 + cg],
          0, 0);
    }
  }
#elif __has_builtin(__builtin_amdgcn_global_load_async_to_lds_b32)
  #pragma unroll
  for (int r = 0; r < 8; ++r) {
    __builtin_amdgcn_global_load_async_to_lds_b32(
        (gint_p)(V + mBase + (size_t)(y0 + r) * kW + x),
        (lint_p)&smask[wib][r * 32 + lane],
        0, 0);
  }
#else
  #pragma unroll
  for (int r = 0; r < 8; ++r)
    smask[wib][r * 32 + lane] = V[mBase + (size_t)(y0 + r) * kW + x];
#endif

#if __has_builtin(__builtin_amdgcn_global_load_async_to_lds_b128) || \
    __has_builtin(__builtin_amdgcn_global_load_async_to_lds_b32)
#if __has_builtin(__builtin_amdgcn_s_wait_asynccnt)
  __builtin_amdgcn_s_wait_asynccnt(0);
#else
  asm volatile("s_wait_asynccnt 0x0" ::: "memory");
#endif
  asm volatile("" ::: "memory");  // keep LDS reads after the async wait
#endif

  // ---- accumulate masked moments; keep s in registers for the output pass ----
  float n = 0.f;
  float St[3]  = {0.f, 0.f, 0.f}, Ss[3]  = {0.f, 0.f, 0.f};
  float Sss[3] = {0.f, 0.f, 0.f}, Sts[3] = {0.f, 0.f, 0.f};
  float sreg[8][3];

  #pragma unroll
  for (int r = 0; r < 8; ++r) {
    const float v = smask[wib][r * 32 + lane];
    n += v;
    const size_t rowOff = (size_t)(y0 + r) * kW + x;
    #pragma unroll
    for (int c = 0; c < 3; ++c) {
      const size_t idx = iBase + (size_t)c * kH * kW + rowOff;
      const float t = __builtin_nontemporal_load(T + idx);
      const float s = __builtin_nontemporal_load(S + idx);
      sreg[r][c] = s;
      const float sv = s * v;
      St[c]  = fmaf(t, v, St[c]);
      Ss[c] += sv;
      Sss[c] = fmaf(s, sv, Sss[c]);
      Sts[c] = fmaf(t, sv, Sts[c]);
    }
  }

  n = red8(n);
  #pragma unroll
  for (int c = 0; c < 3; ++c) {
    St[c]  = red8(St[c]);  Ss[c]  = red8(Ss[c]);
    Sss[c] = red8(Sss[c]); Sts[c] = red8(Sts[c]);
  }

  // ---- per-patch affine solve (all lanes of a group hold identical sums) ----
  const float denom = n + kEPS;
  const float inv   = 1.f / denom;
  const float pv    = (n >= kMinValid) ? 1.f : 0.f;
  float asum = 0.f, bsum = 0.f;
  #pragma unroll
  for (int c = 0; c < 3; ++c) {
    const float tm = St[c] * inv, sm = Ss[c] * inv;
    // centered sums recovered from raw moments (algebraically identical)
    const float svar = (Sss[c] - 2.f * sm * Ss[c] + sm * sm * n) * inv;
    const float cov  = (Sts[c] - tm * Ss[c] - sm * St[c] + tm * sm * n) * inv;
    float alpha = cov / (svar + kEPS);
    float beta  = tm - alpha * sm;
    alpha = alpha * pv + (1.f - pv);  // identity transform if patch invalid
    beta *= pv;
    asum += alpha; bsum += beta;
  }
  const float am = asum * (1.f / 3.f);
  const float bm = bsum * (1.f / 3.f);

  // ---- fused write-back: alpha/beta maps + aligned image, streaming stores ----
  #pragma unroll
  for (int r = 0; r < 8; ++r) {
    const size_t rowOff = (size_t)(y0 + r) * kW + x;
    __builtin_nontemporal_store(am, oAl + mBase + rowOff);
    __builtin_nontemporal_store(bm, oBe + mBase + rowOff);
    #pragma unroll
    for (int c = 0; c < 3; ++c)
      __builtin_nontemporal_store(fmaf(am, sreg[r][c], bm),
                                  oA + iBase + (size_t)c * kH * kW + rowOff);
  }
}

extern "C" void kernel_launch(void* const* d_in, const int* in_sizes, int n_in,
                              void* d_out, int out_size, void* d_ws, size_t ws_size,
                              hipStream_t stream) {
  const float* T = (const float*)d_in[0];  // target_img      (16,3,512,640)
  const float* S = (const float*)d_in[1];  // warped_source   (16,3,512,640)
  const float* V = (const float*)d_in[2];  // valid_mask      (16,1,512,640)
  float* oA  = (float*)d_out;                          // aligned_img
  float* oAl = oA  + (size_t)kB * kC * kH * kW;        // alpha_full
  float* oBe = oAl + (size_t)kB * kH * kW;             // beta_full

  dim3 grid(kW / kBlkCols, kNH, kB);   // (2, 64, 16)
  laa_fused<<<grid, kBlkCols, 0, stream>>>(T, S, V, oA, oAl, oBe);
}